// Lukasiewicz_65970697666731
// MI455X (gfx1250) — compile-verified
//
#include <hip/hip_runtime.h>

// D = relu(X[131072x16] * S[16x696] - bias) as a WMMA f32 GEMM on gfx1250.
// S is the 0/1 subset-selection matrix (singles, pairs, triples of 16 cols,
// lexicographic); bias[j] = |subset(j)| - 1. Column masks are unranked ONCE
// per block into LDS (704 x u32), so the hot loop is: ds_load mask ->
// 4x v_wmma_f32_16x16x4_f32 -> dual-issue bias/relu -> s_clause'd stores.

typedef __attribute__((ext_vector_type(2))) float v2f;  // A/B frag: 16x4 / 4x16 f32
typedef __attribute__((ext_vector_type(8))) float v8f;  // C/D frag: 16x16 f32

#define N_OUT   696     // 16 + C(16,2) + C(16,3) = 16 + 120 + 560
#define N_TILES 44      // ceil(696/16) -> padded to 704
#define N_PAD   (N_TILES * 16)

// 16-bit membership mask of the column subset feeding output column j.
// Only runs during the one-time LDS table fill (<=3 calls per thread).
__device__ __forceinline__ unsigned colMask16(int j) {
    if (j < 16) {                       // identity part
        return 1u << j;
    }
    if (j < 136) {                      // pairs (a,b), lexicographic
        int p = j - 16;
        int a = 0;
        while (p >= 15 - a) { p -= 15 - a; ++a; }
        int b = a + 1 + p;
        return (1u << a) | (1u << b);
    }
    if (j < N_OUT) {                    // triples (a,b,c), lexicographic
        int t = j - 136;
        int a = 0;
        for (;;) {
            int rem = 15 - a;                   // candidates after a
            int cnt = rem * (rem - 1) / 2;      // C(rem, 2)
            if (t < cnt) break;
            t -= cnt; ++a;
        }
        int b = a + 1;
        while (t >= 15 - b) { t -= 15 - b; ++b; }
        int c = b + 1 + t;
        return (1u << a) | (1u << b) | (1u << c);
    }
    return 0u;                          // padding columns (696..703)
}

__global__ __launch_bounds__(256) void
luk_wmma_kernel(const float* __restrict__ x, float* __restrict__ out, int numTilesM) {
    __shared__ unsigned sMask[N_PAD];

    // One-time cooperative fill of the subset-mask table (2816 B of LDS).
    for (int j = threadIdx.x; j < N_PAD; j += 256) {
        sMask[j] = colMask16(j);
    }
    __syncthreads();

    const int lane  = threadIdx.x & 31;
    const int wave  = threadIdx.x >> 5;               // 8 wave32 waves / block
    const int tileM = blockIdx.x * 8 + wave;

    if (tileM < numTilesM) {                          // wave-uniform guard
        const int hi  = lane >> 4;                    // half-wave select
        const int m16 = lane & 15;

        // --- A fragments: 16x16 tile of X, split into four 16x4 K-chunks.
        // Layout per ISA: lane holds M=m16; VGPR v holds K = kbase + v + 2*hi.
        const float* rowp = x + (size_t)(tileM * 16 + m16) * 16;
        v2f a0 = *(const v2f*)(rowp + 0  + 2 * hi);
        v2f a1 = *(const v2f*)(rowp + 4  + 2 * hi);
        v2f a2 = *(const v2f*)(rowp + 8  + 2 * hi);
        v2f a3 = *(const v2f*)(rowp + 12 + 2 * hi);

        for (int tileN = 0; tileN < N_TILES; ++tileN) {
            const int j = tileN * 16 + m16;           // this lane's output column
            const unsigned mask = sMask[j];           // conflict-free ds_load_b32

            v8f acc = {};
#pragma unroll
            for (int kk = 0; kk < 4; ++kk) {
                const int kb = kk * 4 + 2 * hi;
                v2f b;
                b[0] = (mask >> (kb + 0)) & 1u ? 1.0f : 0.0f;
                b[1] = (mask >> (kb + 1)) & 1u ? 1.0f : 0.0f;
                const v2f a = (kk == 0) ? a0 : (kk == 1) ? a1 : (kk == 2) ? a2 : a3;
                // 8 args: (neg_a, A, neg_b, B, c_mod, C, reuse_a, reuse_b)
                acc = __builtin_amdgcn_wmma_f32_16x16x4_f32(
                    false, a, false, b, (short)0, acc, false, false);
            }

            // Lukasiewicz t-norm: max(sum - (|subset|-1), 0)
            const float bias = (float)(__popc(mask) - 1);

            if (j < N_OUT) {
                // C layout: VGPR r, lane L -> row = r + 8*hi, col = L&15.
                // 16 lanes store 64B-contiguous runs per instruction.
                size_t base = (size_t)(tileM * 16 + 8 * hi) * N_OUT + (size_t)j;
#pragma unroll
                for (int r = 0; r < 8; ++r) {
                    out[base + (size_t)r * N_OUT] = fmaxf(acc[r] - bias, 0.0f);
                }
            }
        }
    }
}

extern "C" void kernel_launch(void* const* d_in, const int* in_sizes, int n_in,
                              void* d_out, int out_size, void* d_ws, size_t ws_size,
                              hipStream_t stream) {
    const float* x   = (const float*)d_in[0];
    float*       out = (float*)d_out;

    const int rows      = in_sizes[0] / 16;   // 131072
    const int numTilesM = rows / 16;          // 8192 (rows is a multiple of 16)

    dim3 block(256);                           // 8 wave32 waves
    dim3 grid((numTilesM + 7) / 8);            // one wave per 16-row tile
    luk_wmma_kernel<<<grid, block, 0, stream>>>(x, out, numTilesM);
}